// Kmeans_9131100471960
// MI455X (gfx1250) — compile-verified
//
#include <hip/hip_runtime.h>

// ---------------------------------------------------------------------------
// K-means on MI455X (gfx1250).
// Distance GEMM via V_WMMA_F32_16X16X4_F32:
//   D[m][n] = dot(point_m, mean_n), A = 16 points x 4 dims (C=3 padded),
//   B = meansT (4x16).  argmin_n (|m_n|^2 - 2*D[m][n] + 8) == nearest cluster.
// Argmin across the 16 cluster lanes: pack (score|idx) in u32, VALU-only
// DPP min-reduction within each 16-lane row (no LDS shuffles, no branches).
// ---------------------------------------------------------------------------

typedef __attribute__((ext_vector_type(2))) float v2f;
typedef __attribute__((ext_vector_type(8))) float v8f;

#define KM_N        4194304L
#define KM_K        16
#define KM_ITERS    10
#define NBLK        2048              // assign-kernel grid
#define TILES_PER_WAVE 16             // 16 points per WMMA tile -> 256 pts/wave
#define PTS_PER_BLOCK  (8 * TILES_PER_WAVE * 16)   // 8 waves * 256 = 2048

// Unsigned-min reduction across each 16-lane row using DPP (pure VALU).
// Steps: xor1 (quad_perm 0xB1), xor2 (quad_perm 0x4E),
//        groups-of-4 -> 8 (ROW_HALF_MIRROR 0x141), 8 -> 16 (ROW_MIRROR 0x140).
__device__ __forceinline__ unsigned dpp_rowmin_u32(unsigned u) {
    int x = (int)u, y;
    y = __builtin_amdgcn_update_dpp(x, x, 0x0B1, 0xF, 0xF, true);
    x = (int)min((unsigned)x, (unsigned)y);
    y = __builtin_amdgcn_update_dpp(x, x, 0x04E, 0xF, 0xF, true);
    x = (int)min((unsigned)x, (unsigned)y);
    y = __builtin_amdgcn_update_dpp(x, x, 0x141, 0xF, 0xF, true);
    x = (int)min((unsigned)x, (unsigned)y);
    y = __builtin_amdgcn_update_dpp(x, x, 0x140, 0xF, 0xF, true);
    x = (int)min((unsigned)x, (unsigned)y);
    return (unsigned)x;
}

// ---- copy init means (48 floats) into working buffer ----------------------
__global__ void km_init_kernel(const float* __restrict__ init_means,
                               float* __restrict__ meansA) {
    int t = threadIdx.x;
    if (t < KM_K * 3) meansA[t] = init_means[t];
}

// ---- assignment + per-block partial accumulation --------------------------
__global__ void km_assign_kernel(const float* __restrict__ img,
                                 const float* __restrict__ means,
                                 float* __restrict__ partials) {
    __shared__ float acc[KM_K * 4];   // [k][c0,c1,c2,count]
    int tid = threadIdx.x;
    if (tid < KM_K * 4) acc[tid] = 0.0f;
    __syncthreads();

    int lane = tid & 31;
    int wave = tid >> 5;
    int cl   = lane & 15;          // cluster column owned by this lane in D
    bool hi  = lane >= 16;

    // B-matrix (4x16 = meansT): lanes 0-15 hold rows K=0,1; lanes 16-31 rows K=2,3
    float m0 = means[cl * 3 + 0];
    float m1 = means[cl * 3 + 1];
    float m2 = means[cl * 3 + 2];
    v2f b;
    b.x = hi ? m2 : m0;
    b.y = hi ? 0.0f : m1;
    // +8 shifts scores strictly positive (|2*dot| <= 6) so IEEE bits are
    // monotonic as unsigned -> packed (score|idx) min == argmin w/ tie-break.
    float mn2p8 = m0 * m0 + m1 * m1 + m2 * m2 + 8.0f;

    bool active = (lane < 8) | (lane >= 24);
    int  sel    = lane & 7;

    long base = (long)blockIdx.x * PTS_PER_BLOCK + (long)wave * (TILES_PER_WAVE * 16);

    for (int t = 0; t < TILES_PER_WAVE; ++t) {
        long p0 = base + t * 16;
        // Every lane loads its point's full pixel (uniform, no divergence).
        const float* pp = img + (p0 + cl) * 3;
        float c0 = pp[0], c1 = pp[1], c2 = pp[2];

        // A-matrix (16x4): lane m (0-15) -> (c0,c1); lane 16+m -> (c2, pad)
        v2f a;
        a.x = hi ? c2 : c0;
        a.y = hi ? 0.0f : c1;

        v8f cz = {};
        v8f d = __builtin_amdgcn_wmma_f32_16x16x4_f32(
            /*neg_a=*/false, a, /*neg_b=*/false, b,
            /*c_mod=*/(short)0, cz, /*reuse_a=*/false, /*reuse_b=*/false);

        // D VGPR v: lanes 0-15 = point v vs cluster `lane`; lanes 16-31 = point v+8.
        unsigned r[8];
#pragma unroll
        for (int v = 0; v < 8; ++v) {
            float s = fmaf(-2.0f, d[v], mn2p8);                  // >= 2 > 0
            unsigned u = (__float_as_uint(s) & 0xFFFFFFF0u) | (unsigned)cl;
            r[v] = dpp_rowmin_u32(u);                            // min over 16 lanes
        }

        // lane L in 0..7  -> winner of point L  (lo half of VGPR L)
        // lane 24+v       -> winner of point v+8 (hi half of VGPR v), own pixel too
        unsigned u = r[0];
        u = (sel == 1) ? r[1] : u;
        u = (sel == 2) ? r[2] : u;
        u = (sel == 3) ? r[3] : u;
        u = (sel == 4) ? r[4] : u;
        u = (sel == 5) ? r[5] : u;
        u = (sel == 6) ? r[6] : u;
        u = (sel == 7) ? r[7] : u;
        int w = (int)(u & 15u);

        if (active) {
            atomicAdd(&acc[w * 4 + 0], c0);
            atomicAdd(&acc[w * 4 + 1], c1);
            atomicAdd(&acc[w * 4 + 2], c2);
            atomicAdd(&acc[w * 4 + 3], 1.0f);
        }
    }
    __syncthreads();
    if (tid < KM_K * 4) partials[(long)blockIdx.x * 64 + tid] = acc[tid];
}

// ---- deterministic partial reduction + mean update -------------------------
__global__ void km_update_kernel(const float* __restrict__ partials,
                                 const float* __restrict__ fb,     // 48 floats
                                 float* __restrict__ means_out) {
    __shared__ float red[256];
    int tid  = threadIdx.x;
    int slot = tid & 63;
    int part = tid >> 6;           // 0..3
    float s = 0.0f;
    for (int blk = part; blk < NBLK; blk += 4)
        s += partials[(long)blk * 64 + slot];
    red[tid] = s;
    __syncthreads();
    if (tid < 64)
        red[tid] = red[tid] + red[tid + 64] + red[tid + 128] + red[tid + 192];
    __syncthreads();
    if (tid < KM_K * 3) {
        int k = tid / 3, ch = tid % 3;
        float cnt = red[k * 4 + 3];
        means_out[tid] = (cnt > 0.0f) ? (red[k * 4 + ch] / cnt) : fb[tid];
    }
}

// ---- final reconstruction: r_final via means_9, gather from means_10 -------
__global__ void km_recon_kernel(const float* __restrict__ img,
                                const float* __restrict__ means_assign,  // means_9
                                const float* __restrict__ means_final,   // means_10
                                float* __restrict__ out, long n) {
    __shared__ float ma[KM_K * 3];
    __shared__ float mf[KM_K * 3];
    int tid = threadIdx.x;
    if (tid < KM_K * 3) { ma[tid] = means_assign[tid]; mf[tid] = means_final[tid]; }
    __syncthreads();

    long i = (long)blockIdx.x * blockDim.x + tid;
    float x0 = img[i * 3 + 0];
    float x1 = img[i * 3 + 1];
    float x2 = img[i * 3 + 2];
    float best = 3.4e38f;
    int w = 0;
#pragma unroll
    for (int k = 0; k < KM_K; ++k) {
        float d0 = x0 - ma[k * 3 + 0];
        float d1 = x1 - ma[k * 3 + 1];
        float d2 = x2 - ma[k * 3 + 2];
        float s = d0 * d0 + d1 * d1 + d2 * d2;
        bool take = s < best;                 // strict '<' == first-index tie-break
        best = take ? s : best;
        w    = take ? k : w;
    }
    out[i * 3 + 0] = mf[w * 3 + 0] * 255.0f;
    out[i * 3 + 1] = mf[w * 3 + 1] * 255.0f;
    out[i * 3 + 2] = mf[w * 3 + 2] * 255.0f;

    if (blockIdx.x == 0 && tid < KM_K * 3)
        out[n * 3 + tid] = mf[tid];          // append final means to d_out tail
}

// ---------------------------------------------------------------------------
extern "C" void kernel_launch(void* const* d_in, const int* in_sizes, int n_in,
                              void* d_out, int out_size, void* d_ws, size_t ws_size,
                              hipStream_t stream) {
    const float* img        = (const float*)d_in[0];  // [N,3]
    const float* init_means = (const float*)d_in[1];  // [16,3]
    const float* fallback   = (const float*)d_in[2];  // [10,16,3]
    float* out = (float*)d_out;

    float* ws       = (float*)d_ws;
    float* meansA   = ws;            // 64 floats (48 used)
    float* meansB   = ws + 64;       // 64 floats
    float* partials = ws + 128;      // NBLK * 64 floats (~512 KB)

    km_init_kernel<<<1, 64, 0, stream>>>(init_means, meansA);

    for (int it = 0; it < KM_ITERS; ++it) {
        float* cur = (it & 1) ? meansB : meansA;
        float* nxt = (it & 1) ? meansA : meansB;
        km_assign_kernel<<<NBLK, 256, 0, stream>>>(img, cur, partials);
        km_update_kernel<<<1, 256, 0, stream>>>(partials, fallback + it * 48, nxt);
    }
    // after it=9: meansB holds means_9 (assignment means), meansA holds means_10
    km_recon_kernel<<<(int)(KM_N / 256), 256, 0, stream>>>(img, meansB, meansA, out, KM_N);
}